// GCNEncoder_Multiinput_35785667510355
// MI455X (gfx1250) — compile-verified
//
#include <hip/hip_runtime.h>

typedef float v2f __attribute__((ext_vector_type(2)));
typedef float v8f __attribute__((ext_vector_type(8)));

#define NNODES 100000
#define NEDGES 1600000
#define DIN    256
#define DOUT   128
#define NHEAD  4
#define CHAN   32
#define EPSVAL 1e-5f

// workspace layout (float offsets)
#define OFF_Q   0L
#define OFF_K   12800000L
#define OFF_V   25600000L
#define OFF_SC  38400000L                 // E*H  = 6,400,000
#define OFF_MK  44800000L                 // N*H  encoded-max keys (uint)
#define OFF_DEN 45200000L                 // N*H  softmax denominators
#define OFF_CS  45600000L                 // 128 column sums
#define OFF_CSQ 45600128L                 // 128 column sumsq
#define ZERO_CNT 800256L                  // mkey+denom+cs+csq

// ---------------- monotone float<->uint encoding for atomicMax ----------
__device__ __forceinline__ unsigned fenc(float f) {
  unsigned u = __float_as_uint(f);
  return (u & 0x80000000u) ? ~u : (u | 0x80000000u);
}
__device__ __forceinline__ float fdec(unsigned kk) {
  return (kk & 0x80000000u) ? __uint_as_float(kk ^ 0x80000000u)
                            : __uint_as_float(~kk);
}

// ---------------- 0: zero the atomic-accumulator region ------------------
__global__ __launch_bounds__(256) void ws_zero(unsigned* __restrict__ p, long n) {
  long t = (long)blockIdx.x * 256 + threadIdx.x;
  if (t < n) p[t] = 0u;
}

// ---------------- 1: fused q/k/v/skip GEMM via V_WMMA_F32_16X16X4_F32 ----
// workgroup: 256 threads = 8 waves, 32 rows of x staged in LDS.
// concat column space: [0,128)=q [128,256)=k [256,384)=v [384,512)=skip
__global__ __launch_bounds__(256)
void qkvs_gemm(const float* __restrict__ x,
               const float* __restrict__ Wq, const float* __restrict__ Wk,
               const float* __restrict__ Wv, const float* __restrict__ Wsk,
               const float* __restrict__ bq, const float* __restrict__ bk,
               const float* __restrict__ bv, const float* __restrict__ bsk,
               float* __restrict__ qo, float* __restrict__ ko,
               float* __restrict__ vo, float* __restrict__ so)
{
  __shared__ float xs[32 * 260];               // 260-float padded stride
  const int tid    = threadIdx.x;
  const int wv     = tid >> 5;
  const int lane   = tid & 31;
  const int lane16 = lane & 15;
  const int khalf  = lane >> 4;                // 0: K{0,1}  1: K{2,3}
  const long r0    = (long)blockIdx.x * 32;    // N == 3125*32 exactly

  // stage 32 contiguous rows of x (32KB) into LDS with padding
  for (int i = tid; i < 32 * 64; i += 256) {
    int row = i >> 6;
    int c4  = (i & 63) << 2;
    float4 val = *(const float4*)(x + (r0 + row) * DIN + c4);
    *(float4*)(&xs[row * 260 + c4]) = val;
  }
  __syncthreads();

  const float* arow0 = xs + lane16 * 260;          // row tile 0: M = lane16 (+8)
  const float* arow1 = xs + (16 + lane16) * 260;   // row tile 1

  for (int ct = wv; ct < 32; ct += 8) {            // 4 column tiles per wave
    const int colg = ct << 4;
    const int mi   = colg >> 7;
    const int cc   = colg & 127;
    const float* W   = (mi == 0) ? Wq : (mi == 1) ? Wk : (mi == 2) ? Wv : Wsk;
    const float* b   = (mi == 0) ? bq : (mi == 1) ? bk : (mi == 2) ? bv : bsk;
    float*       op  = (mi == 0) ? qo : (mi == 1) ? ko : (mi == 2) ? vo : so;
    const int nc = cc + lane16;
    const float bias = b[nc];

    v8f acc0, acc1;
#pragma unroll
    for (int j = 0; j < 8; ++j) { acc0[j] = bias; acc1[j] = bias; }

    const float* wcol = W + nc;                  // W is [256,128] row-major
#pragma unroll 4
    for (int k0 = 0; k0 < DIN; k0 += 4) {
      const int kk = k0 + khalf * 2;
      // A fragments per ISA 16x4 f32 layout (8B-aligned ds_load_b64)
      v2f a0 = *(const v2f*)(arow0 + kk);
      v2f a1 = *(const v2f*)(arow1 + kk);
      // B fragment: 4x16, VGPR0 = row kk, VGPR1 = row kk+1, N across lanes
      v2f bm;
      bm.x = wcol[(long)kk * DOUT];
      bm.y = wcol[(long)(kk + 1) * DOUT];
      acc0 = __builtin_amdgcn_wmma_f32_16x16x4_f32(false, a0, false, bm,
                                                   (short)0, acc0, false, false);
      acc1 = __builtin_amdgcn_wmma_f32_16x16x4_f32(false, a1, false, bm,
                                                   (short)0, acc1, false, false);
    }

    // C/D layout: VGPR j -> M=j (lanes 0-15) / M=j+8 (lanes 16-31), N=lane16
    const int rowoff = khalf * 8;
#pragma unroll
    for (int j = 0; j < 8; ++j) {
      long row0 = r0 + j + rowoff;
      long row1 = r0 + 16 + j + rowoff;
      op[row0 * DOUT + nc] = acc0[j];
      op[row1 * DOUT + nc] = acc1[j];
    }
  }
}

// ---------------- 2: per-edge attention logits + segment max -------------
__global__ __launch_bounds__(256)
void edge_score(const int* __restrict__ ei, const float* __restrict__ q,
                const float* __restrict__ k, float* __restrict__ score,
                unsigned* __restrict__ mkey)
{
  long t = (long)blockIdx.x * 256 + threadIdx.x;
  if (t >= (long)NEDGES * NHEAD) return;
  int e = (int)(t >> 2);
  int h = (int)(t & 3);
  int src = ei[e];
  int dst = ei[NEDGES + e];
  const float4* qr = (const float4*)(q + (long)dst * DOUT + h * CHAN);
  const float4* kr = (const float4*)(k + (long)src * DOUT + h * CHAN);
  float s = 0.f;
#pragma unroll
  for (int i = 0; i < 8; ++i) {
    float4 a = qr[i]; float4 b = kr[i];
    s += a.x * b.x + a.y * b.y + a.z * b.z + a.w * b.w;
  }
  s *= 0.17677669529663687f;            // 1/sqrt(32)
  score[t] = s;
  atomicMax(&mkey[(long)dst * NHEAD + h], fenc(s));
}

// ---------------- 3: exp(score - max) + segment sum ----------------------
__global__ __launch_bounds__(256)
void edge_expsum(const int* __restrict__ ei, float* __restrict__ score,
                 const unsigned* __restrict__ mkey, float* __restrict__ denom)
{
  long t = (long)blockIdx.x * 256 + threadIdx.x;
  if (t >= (long)NEDGES * NHEAD) return;
  int e = (int)(t >> 2);
  int h = (int)(t & 3);
  int dst = ei[NEDGES + e];
  float mx = fdec(mkey[(long)dst * NHEAD + h]);
  float ex = __expf(score[t] - mx);
  score[t] = ex;
  atomicAdd(&denom[(long)dst * NHEAD + h], ex);
}

// ---------------- 4: alpha-weighted aggregation of v ---------------------
// one thread per (edge, 4 channels): 32 threads cover one edge's 128 outputs
__global__ __launch_bounds__(256)
void edge_aggregate(const int* __restrict__ ei, const float* __restrict__ esc,
                    const float* __restrict__ denom, const float* __restrict__ v,
                    float* __restrict__ out)
{
  long t = (long)blockIdx.x * 256 + threadIdx.x;
  if (t >= (long)NEDGES * 32) return;
  int e  = (int)(t >> 5);
  int r  = (int)(t & 31);
  int h  = r >> 3;
  int c4 = (r & 7) << 2;
  int src = ei[e];
  int dst = ei[NEDGES + e];
  float alpha = esc[(long)e * NHEAD + h] / denom[(long)dst * NHEAD + h];
  const float4 vv = *(const float4*)(v + (long)src * DOUT + h * CHAN + c4);
  float* op = out + (long)dst * DOUT + h * CHAN + c4;
  atomicAdd(op + 0, alpha * vv.x);
  atomicAdd(op + 1, alpha * vv.y);
  atomicAdd(op + 2, alpha * vv.z);
  atomicAdd(op + 3, alpha * vv.w);
}

// ---------------- 5: per-column sum / sumsq ------------------------------
__global__ __launch_bounds__(128)
void col_stats(const float* __restrict__ out, float* __restrict__ csum,
               float* __restrict__ csq)
{
  int col = threadIdx.x;
  long rbeg = (long)blockIdx.x * 128;
  long rend = rbeg + 128;
  if (rend > (long)NNODES) rend = NNODES;
  float s = 0.f, s2 = 0.f;
  for (long r = rbeg; r < rend; ++r) {
    float vx = out[r * DOUT + col];
    s += vx; s2 += vx * vx;
  }
  atomicAdd(&csum[col], s);
  atomicAdd(&csq[col],  s2);
}

// ---------------- 6: GraphNorm + Mish (fused, in place) ------------------
__global__ __launch_bounds__(256)
void norm_mish(float* __restrict__ out, const float* __restrict__ csum,
               const float* __restrict__ csq, const float* __restrict__ gw,
               const float* __restrict__ gb, const float* __restrict__ gms)
{
  long t = (long)blockIdx.x * 256 + threadIdx.x;
  if (t >= (long)NNODES * DOUT) return;
  int col = (int)(t & 127);
  const float invN = 1.0f / (float)NNODES;
  float mu  = csum[col] * invN;
  float ex2 = csq[col] * invN;
  float c   = mu * gms[col];
  float var = ex2 - 2.f * c * mu + c * c;       // E[(x-c)^2]
  float cent = out[t] - c;
  float y = gw[col] * cent * rsqrtf(var + EPSVAL) + gb[col];
  float sp = (y > 20.f) ? y : log1pf(__expf(y));  // softplus
  out[t] = y * tanhf(sp);                         // mish
}

// ---------------- launcher ----------------------------------------------
extern "C" void kernel_launch(void* const* d_in, const int* in_sizes, int n_in,
                              void* d_out, int out_size, void* d_ws, size_t ws_size,
                              hipStream_t stream) {
  const float* x   = (const float*)d_in[0];
  const int*   ei  = (const int*)  d_in[1];
  const float* Wq  = (const float*)d_in[2];
  const float* bq  = (const float*)d_in[3];
  const float* Wk  = (const float*)d_in[4];
  const float* bk  = (const float*)d_in[5];
  const float* Wv  = (const float*)d_in[6];
  const float* bv  = (const float*)d_in[7];
  const float* Wsk = (const float*)d_in[8];
  const float* bsk = (const float*)d_in[9];
  const float* gw  = (const float*)d_in[10];
  const float* gb  = (const float*)d_in[11];
  const float* gms = (const float*)d_in[12];

  float* out = (float*)d_out;
  float* ws  = (float*)d_ws;
  float*    qbuf = ws + OFF_Q;
  float*    kbuf = ws + OFF_K;
  float*    vbuf = ws + OFF_V;
  float*    sc   = ws + OFF_SC;
  unsigned* mkey = (unsigned*)(ws + OFF_MK);
  float*    den  = ws + OFF_DEN;
  float*    cs   = ws + OFF_CS;
  float*    csq  = ws + OFF_CSQ;

  // zero atomic accumulators (mkey identity == 0 under fenc ordering)
  ws_zero<<<(ZERO_CNT + 255) / 256, 256, 0, stream>>>(mkey, ZERO_CNT);

  // fused q/k/v/skip GEMM; skip term written straight into d_out
  qkvs_gemm<<<NNODES / 32, 256, 0, stream>>>(x, Wq, Wk, Wv, Wsk,
                                             bq, bk, bv, bsk,
                                             qbuf, kbuf, vbuf, out);

  // segment softmax over incoming edges
  edge_score<<<(NEDGES * NHEAD) / 256, 256, 0, stream>>>(ei, qbuf, kbuf, sc, mkey);
  edge_expsum<<<(NEDGES * NHEAD) / 256, 256, 0, stream>>>(ei, sc, mkey, den);
  edge_aggregate<<<(NEDGES * 32) / 256, 256, 0, stream>>>(ei, sc, den, vbuf, out);

  // GraphNorm statistics + fused normalize/Mish
  col_stats<<<(NNODES + 127) / 128, 128, 0, stream>>>(out, cs, csq);
  norm_mish<<<(NNODES * DOUT) / 256, 256, 0, stream>>>(out, cs, csq, gw, gb, gms);
}